// Layer_3DPE_attention_86878598464032
// MI455X (gfx1250) — compile-verified
//
#include <hip/hip_runtime.h>

typedef __attribute__((ext_vector_type(2))) float v2f;
typedef __attribute__((ext_vector_type(8))) float v8f;

#define WMMA_F32(a, b, c) \
    __builtin_amdgcn_wmma_f32_16x16x4_f32(false, (a), false, (b), (short)0, (c), false, false)

constexpr int BB   = 16;
constexpr int DIN  = 64;
constexpr int DOUT = 64;
constexpr int P1D  = 64;
constexpr int P2D  = 32;
constexpr int P3D  = 16;
constexpr int M512 = P2D * P3D;        // 512 = flattened (p2*p3)
constexpr int NPB  = P1D * M512;       // 32768 elements per (b, channel)
constexpr long long TOT = (long long)BB * DOUT * NPB;   // 33,554,432

// ---------------------------------------------------------------------------
// Mean over z (last dim, 16): Amz[b,i,x,y] = mean_z A[b,i,x,y,z]
// ---------------------------------------------------------------------------
__global__ void meanz_kernel(const float* __restrict__ A, float* __restrict__ Amz) {
    long long t = (long long)blockIdx.x * blockDim.x + threadIdx.x;   // 2,097,152
    const float4* p = (const float4*)(A + t * 16);
    float4 s0 = p[0], s1 = p[1], s2 = p[2], s3 = p[3];
    float s = (s0.x + s0.y + s0.z + s0.w) + (s1.x + s1.y + s1.z + s1.w) +
              (s2.x + s2.y + s2.z + s2.w) + (s3.x + s3.y + s3.z + s3.w);
    Amz[t] = s * (1.0f / 16.0f);
}

// ---------------------------------------------------------------------------
// Mean over y (dim p2=32): Amy[b,i,x,z] = mean_y A[b,i,x,y,z]
// ---------------------------------------------------------------------------
__global__ void meany_kernel(const float* __restrict__ A, float* __restrict__ Amy) {
    long long t = (long long)blockIdx.x * blockDim.x + threadIdx.x;   // 1,048,576
    long long bix = t >> 4;
    int z = (int)(t & 15);
    const float* p = A + bix * M512 + z;
    float s = 0.f;
#pragma unroll
    for (int y = 0; y < 32; ++y) s += p[y * 16];
    Amy[t] = s * (1.0f / 32.0f);
}

// ---------------------------------------------------------------------------
// Channel-projection GEMM: O_w[b,o,n] = sum_i W_w[o,i] * Ain[b,i,n]
// One wave per 16-wide n-tile; all 4 M-tiles (64 outs) and NW weights held
// as accumulators so each data B-fragment feeds 4*NW WMMAs.
// ---------------------------------------------------------------------------
template <int NW>
__global__ void proj_gemm_kernel(const float* __restrict__ Ain,
                                 const float* __restrict__ W0, const float* __restrict__ W1,
                                 const float* __restrict__ W2, const float* __restrict__ W3,
                                 float* __restrict__ O0, float* __restrict__ O1,
                                 float* __restrict__ O2, float* __restrict__ O3, int N) {
    const float* Ws[4] = {W0, W1, W2, W3};
    float*       Os[4] = {O0, O1, O2, O3};
    const int lane = threadIdx.x & 31;
    const int wv   = threadIdx.x >> 5;
    const int ln   = lane & 15;
    const int hi   = lane >> 4;
    const int nt   = blockIdx.x * (blockDim.x >> 5) + wv;
    const int b    = blockIdx.y;
    const int n0   = nt * 16;
    const float* Ab = Ain + (long long)b * DIN * N;

    v8f acc[NW][4];
#pragma unroll
    for (int w = 0; w < NW; ++w)
#pragma unroll
        for (int mo = 0; mo < 4; ++mo) acc[w][mo] = (v8f){0.f, 0.f, 0.f, 0.f, 0.f, 0.f, 0.f, 0.f};

    for (int kk = 0; kk < 16; ++kk) {            // K = DIN = 64, 4 per WMMA
        const int kr = kk * 4 + 2 * hi;
        v2f bf;
        bf.x = Ab[(long long)kr * N + n0 + ln];
        bf.y = Ab[(long long)(kr + 1) * N + n0 + ln];
#pragma unroll
        for (int w = 0; w < NW; ++w) {
#pragma unroll
            for (int mo = 0; mo < 4; ++mo) {
                v2f af = *(const v2f*)(Ws[w] + (mo * 16 + ln) * DIN + kr);
                acc[w][mo] = WMMA_F32(af, bf, acc[w][mo]);
            }
        }
    }

#pragma unroll
    for (int w = 0; w < NW; ++w) {
        float* Op = Os[w] + (long long)b * DOUT * N;
#pragma unroll
        for (int mo = 0; mo < 4; ++mo) {
#pragma unroll
            for (int r = 0; r < 8; ++r) {
                const int m = mo * 16 + r + 8 * hi;
                Op[(long long)m * N + n0 + ln] = acc[w][mo][r];
            }
        }
    }
}

// ---------------------------------------------------------------------------
// Alpha = Q*K^T per (b,c), then Alpha_norm = |Alpha| / ||row||_2.
// Block = one (b,c); 8 waves cover the 4x4 tile grid (2 tiles/wave, shared A).
// ---------------------------------------------------------------------------
__global__ void alpha_kernel(const float* __restrict__ Qb, const float* __restrict__ Kb,
                             float* __restrict__ AlphaN) {
    __shared__ float rowsq[64][2];
    const int bi   = blockIdx.x;                 // b*64 + c
    const int lane = threadIdx.x & 31;
    const int wv   = threadIdx.x >> 5;
    const int ln   = lane & 15;
    const int hi   = lane >> 4;
    const int xo   = wv >> 1;                    // M-tile (rows of Alpha)
    const int pp   = wv & 1;                     // which yo-pair
    const float* Q = Qb + (long long)bi * NPB;   // [64][512]
    const float* K = Kb + (long long)bi * NPB;   // [64][512]
    const int y0a  = (pp * 2 + 0) * 16;
    const int y0b  = (pp * 2 + 1) * 16;

    v8f a0 = (v8f){0.f, 0.f, 0.f, 0.f, 0.f, 0.f, 0.f, 0.f};
    v8f a1 = a0;

    for (int kk = 0; kk < 128; ++kk) {           // K = 512, 4 per WMMA
        const int kr = kk * 4 + 2 * hi;
        v2f af = *(const v2f*)(Q + (xo * 16 + ln) * M512 + kr);
        v2f b0 = *(const v2f*)(K + (y0a + ln) * M512 + kr);   // B = K^T
        v2f b1 = *(const v2f*)(K + (y0b + ln) * M512 + kr);
        a0 = WMMA_F32(af, b0, a0);
        a1 = WMMA_F32(af, b1, a1);
    }

    // Row-wise sum of squares: butterfly within each 16-lane half, then LDS
    // combine of exactly two wave partials per row (commutative -> deterministic).
#pragma unroll
    for (int r = 0; r < 8; ++r) {
        float p = a0[r] * a0[r] + a1[r] * a1[r];
        p += __shfl_xor(p, 1, 16);
        p += __shfl_xor(p, 2, 16);
        p += __shfl_xor(p, 4, 16);
        p += __shfl_xor(p, 8, 16);
        if (ln == 0) rowsq[xo * 16 + r + 8 * hi][pp] = p;
    }
    __syncthreads();

    float* An = AlphaN + (long long)bi * 4096;   // [64][64]
#pragma unroll
    for (int r = 0; r < 8; ++r) {
        const int x = xo * 16 + r + 8 * hi;
        const float rn = rsqrtf(rowsq[x][0] + rowsq[x][1]);
        An[x * 64 + y0a + ln] = fabsf(a0[r]) * rn;
        An[x * 64 + y0b + ln] = fabsf(a1[r]) * rn;
    }
}

// ---------------------------------------------------------------------------
// Y = Alpha_norm * V per (b,c); fuse out = relu(A1 + 0.1*(A2 + A3 + Y)),
// store preBN in place over A1, accumulate per-(b,c) sum / sumsq for BN.
// ---------------------------------------------------------------------------
__global__ void ycombine_kernel(const float* __restrict__ AlphaN, const float* __restrict__ Vb,
                                const float* __restrict__ A2, const float* __restrict__ A3,
                                float* __restrict__ A1pre, float* __restrict__ bnSum,
                                float* __restrict__ bnSq) {
    __shared__ float ssum[8], ssq[8];
    const int bi   = blockIdx.x;                 // b*64 + c
    const int lane = threadIdx.x & 31;
    const int wv   = threadIdx.x >> 5;
    const int ln   = lane & 15;
    const int hi   = lane >> 4;
    const float* An  = AlphaN + (long long)bi * 4096;          // [64][64]
    const float* V   = Vb + (long long)bi * NPB;               // [64][512]
    const float* A2p = A2 + (long long)bi * (P1D * P2D);       // [64][32]
    const float* A3p = A3 + (long long)bi * (P1D * P3D);       // [64][16]
    float* Pre = A1pre + (long long)bi * NPB;                  // [64][512]

    float lsum = 0.f, lsq = 0.f;

    for (int t = 0; t < 4; ++t) {
        const int nt = wv * 4 + t;               // 32 n-tiles over m=512
        const int m0 = nt * 16;
        v8f acc[4];
#pragma unroll
        for (int xo = 0; xo < 4; ++xo) acc[xo] = (v8f){0.f, 0.f, 0.f, 0.f, 0.f, 0.f, 0.f, 0.f};

        for (int kk = 0; kk < 16; ++kk) {        // K = 64 (y), 4 per WMMA
            const int kr = kk * 4 + 2 * hi;
            v2f bf;
            bf.x = V[kr * M512 + m0 + ln];
            bf.y = V[(kr + 1) * M512 + m0 + ln];
#pragma unroll
            for (int xo = 0; xo < 4; ++xo) {
                v2f af = *(const v2f*)(An + (xo * 16 + ln) * 64 + kr);
                acc[xo] = WMMA_F32(af, bf, acc[xo]);
            }
        }

        const int m  = m0 + ln;
        const int y2 = m >> 4;                   // p2 index
        const int z  = m & 15;                   // p3 index
#pragma unroll
        for (int xo = 0; xo < 4; ++xo) {
#pragma unroll
            for (int r = 0; r < 8; ++r) {
                const int x = xo * 16 + r + 8 * hi;
                float v = Pre[x * M512 + m] +
                          0.1f * (A2p[x * P2D + y2] + A3p[x * P3D + z] + acc[xo][r]);
                v = fmaxf(v, 0.f);
                Pre[x * M512 + m] = v;
                lsum += v;
                lsq  += v * v;
            }
        }
    }

    // Deterministic block reduction: fixed butterfly per wave, fixed-order sum.
#pragma unroll
    for (int msk = 1; msk < 32; msk <<= 1) {
        lsum += __shfl_xor(lsum, msk, 32);
        lsq  += __shfl_xor(lsq, msk, 32);
    }
    if (lane == 0) { ssum[wv] = lsum; ssq[wv] = lsq; }
    __syncthreads();
    if (threadIdx.x == 0) {
        float S = 0.f, Qs = 0.f;
        for (int w = 0; w < 8; ++w) { S += ssum[w]; Qs += ssq[w]; }
        bnSum[bi] = S;
        bnSq[bi]  = Qs;
    }
}

// ---------------------------------------------------------------------------
// Per-channel BN statistics (fixed-order sum over b -> deterministic).
// ---------------------------------------------------------------------------
__global__ void bnstats_kernel(const float* __restrict__ bnSum, const float* __restrict__ bnSq,
                               const float* __restrict__ gamma, const float* __restrict__ beta,
                               float* __restrict__ scale, float* __restrict__ shift) {
    const int c = threadIdx.x;
    if (c < DOUT) {
        float S = 0.f, Q = 0.f;
        for (int b = 0; b < BB; ++b) { S += bnSum[b * DOUT + c]; Q += bnSq[b * DOUT + c]; }
        const float invN = 1.0f / ((float)BB * (float)NPB);
        const float mean = S * invN;
        const float var  = Q * invN - mean * mean;
        const float sc   = gamma[c] * rsqrtf(var + 1e-5f);
        scale[c] = sc;
        shift[c] = beta[c] - mean * sc;
    }
}

// ---------------------------------------------------------------------------
// Apply BN: out = preBN * scale[c] + shift[c], float4-vectorized.
// ---------------------------------------------------------------------------
__global__ void bnapply_kernel(const float* __restrict__ Pre, const float* __restrict__ scale,
                               const float* __restrict__ shift, float* __restrict__ out) {
    const long long e0 = ((long long)blockIdx.x * blockDim.x + threadIdx.x) * 4;
    const int c = (int)((e0 >> 15) & 63);        // element -> channel (32768 per (b,c))
    const float4 p = *(const float4*)(Pre + e0);
    const float s = scale[c], sh = shift[c];
    float4 o;
    o.x = p.x * s + sh;
    o.y = p.y * s + sh;
    o.z = p.z * s + sh;
    o.w = p.w * s + sh;
    *(float4*)(out + e0) = o;
}

// ---------------------------------------------------------------------------
extern "C" void kernel_launch(void* const* d_in, const int* in_sizes, int n_in,
                              void* d_out, int out_size, void* d_ws, size_t ws_size,
                              hipStream_t stream) {
    (void)in_sizes; (void)n_in; (void)out_size; (void)ws_size;
    const float* A     = (const float*)d_in[0];
    const float* P1    = (const float*)d_in[1];
    const float* P2    = (const float*)d_in[2];
    const float* P3    = (const float*)d_in[3];
    const float* P4    = (const float*)d_in[4];
    const float* qw    = (const float*)d_in[5];
    const float* kw    = (const float*)d_in[6];
    const float* gamma = (const float*)d_in[7];
    const float* beta  = (const float*)d_in[8];

    float* ws = (float*)d_ws;
    const long long SZ = TOT;                    // 33,554,432 floats per big buffer
    float* A1  = ws;                             // P1 proj -> later preBN (in place)
    float* Qb  = ws + SZ;
    float* Kb  = ws + 2 * SZ;
    float* Vb  = ws + 3 * SZ;
    float* Amz = ws + 4 * SZ;                    // 2,097,152
    float* Amy = Amz + 2097152;                  // 1,048,576
    float* A2  = Amy + 1048576;                  // 2,097,152
    float* A3  = A2 + 2097152;                   // 1,048,576
    float* AlN = A3 + 1048576;                   // 4,194,304
    float* bnS = AlN + 4194304;                  // 1,024
    float* bnQ = bnS + 1024;                     // 1,024
    float* bnScale = bnQ + 1024;                 // 64
    float* bnShift = bnScale + 64;               // 64

    meanz_kernel<<<2097152 / 256, 256, 0, stream>>>(A, Amz);
    meany_kernel<<<1048576 / 256, 256, 0, stream>>>(A, Amy);

    // Main 4-weight projection over A: A1 = P1*A, Q = q*A, K = k*A, V = P4*A
    proj_gemm_kernel<4><<<dim3(NPB / 64, BB), 128, 0, stream>>>(
        A, P1, qw, kw, P4, A1, Qb, Kb, Vb, NPB);
    // A2 = P2 * mean_z(A)   (N = 64*32 = 2048 per b)
    proj_gemm_kernel<1><<<dim3(2048 / 64, BB), 128, 0, stream>>>(
        Amz, P2, nullptr, nullptr, nullptr, A2, nullptr, nullptr, nullptr, 2048);
    // A3 = P3 * mean_y(A)   (N = 64*16 = 1024 per b)
    proj_gemm_kernel<1><<<dim3(1024 / 64, BB), 128, 0, stream>>>(
        Amy, P3, nullptr, nullptr, nullptr, A3, nullptr, nullptr, nullptr, 1024);

    alpha_kernel<<<BB * DOUT, 256, 0, stream>>>(Qb, Kb, AlN);
    ycombine_kernel<<<BB * DOUT, 256, 0, stream>>>(AlN, Vb, A2, A3, A1, bnS, bnQ);
    bnstats_kernel<<<1, 64, 0, stream>>>(bnS, bnQ, gamma, beta, bnScale, bnShift);
    bnapply_kernel<<<(int)(SZ / 4 / 256), 256, 0, stream>>>(A1, bnScale, bnShift, (float*)d_out);
}